// ConvPool_81819126988920
// MI455X (gfx1250) — compile-verified
//
#include <hip/hip_runtime.h>
#include <hip/hip_bf16.h>

// GCNConv: out[i] = sum_{j->i} dinv[j]*dinv[i] * (x[j] @ W) + b, self-loops added,
// deg = in-degree(col) + 1.
//
// Phases:
//   K0 init:    out = broadcast(b); deg = 1.0 (self loop)
//   K1 degree:  atomicAdd(deg[col[e]], 1)
//   K2 rsqrt:   dinv = rsqrt(deg)  (in place)
//   K3 gemm:    h = x @ W  via V_WMMA_F32_16X16X4_F32 (fp32-exact)
//   K4 scatter: wave-per-edge: out[col] += (dinv[row]*dinv[col]) * h[row]
//               (+ N virtual self-loop edges row==col)

typedef float v2f __attribute__((ext_vector_type(2)));
typedef float v8f __attribute__((ext_vector_type(8)));

#define D 128  // D_IN == D_OUT == 128

__global__ void gcn_init(float* __restrict__ out, float* __restrict__ deg,
                         const float* __restrict__ b, int N) {
  int idx = blockIdx.x * blockDim.x + threadIdx.x;
  if (idx < N * D) out[idx] = b[idx & (D - 1)];
  if (idx < N) deg[idx] = 1.0f;  // self-loop contribution
}

__global__ void gcn_degree(const long long* __restrict__ ei, float* __restrict__ deg, int E) {
  int e = blockIdx.x * blockDim.x + threadIdx.x;
  if (e < E) {
    int col = (int)ei[E + e];  // edge_index[1][e] = destination
    atomicAdd(&deg[col], 1.0f);
  }
}

__global__ void gcn_rsqrt(float* __restrict__ deg, int N) {
  int i = blockIdx.x * blockDim.x + threadIdx.x;
  if (i < N) {
    float d = deg[i];
    deg[i] = (d > 0.0f) ? rsqrtf(d) : 0.0f;  // now holds dinv
  }
}

// h = x @ W. One 256-thread block (8 waves) covers a 16-row stripe; wave w
// produces the 16x16 C tile at columns [16w, 16w+16). K marches in steps of 4
// with V_WMMA_F32_16X16X4_F32 (A: 16x4 f32 in 2 VGPRs, B: 4x16 f32 in 2 VGPRs,
// C/D: 16x16 f32 in 8 VGPRs).
__global__ void __launch_bounds__(256) gcn_gemm(const float* __restrict__ x,
                                                const float* __restrict__ W,
                                                float* __restrict__ h, int N) {
  const int wave = threadIdx.x >> 5;   // 0..7 -> column tile
  const int lane = threadIdx.x & 31;
  const int half = lane >> 4;          // 0: K={0,1}, 1: K={2,3}
  const int l    = lane & 15;          // M (for A) / N (for B) within tile
  const int m0 = blockIdx.x * 16;
  const int n0 = wave * 16;
  const int koff = half * 2;

  int mrow = m0 + l;
  if (mrow > N - 1) mrow = N - 1;      // branchless clamp (keeps EXEC all-ones)
  const float* __restrict__ xrow = x + (size_t)mrow * D;

  v8f c = {};
#pragma unroll
  for (int k = 0; k < D; k += 4) {
    v2f a;
    a.x = xrow[k + koff];
    a.y = xrow[k + koff + 1];
    v2f bf;
    bf.x = W[(size_t)(k + koff) * D + n0 + l];
    bf.y = W[(size_t)(k + koff + 1) * D + n0 + l];
    // (neg_a, A, neg_b, B, c_mod, C, reuse_a, reuse_b)
    c = __builtin_amdgcn_wmma_f32_16x16x4_f32(false, a, false, bf,
                                              (short)0, c, false, false);
  }

#pragma unroll
  for (int g = 0; g < 8; ++g) {
    int rr = m0 + half * 8 + g;        // C/D layout: lanes16-31 carry M=8..15
    if (rr < N) h[(size_t)rr * D + n0 + l] = c[g];
  }
}

// One wave per (edge or self-loop). Lane L owns float4 at column 4L of the
// 128-wide feature row: coalesced 512B gather from h[row], 4 f32 atomics to
// out[col].
__global__ void gcn_scatter(const long long* __restrict__ ei,
                            const float* __restrict__ h,
                            const float* __restrict__ dinv,
                            float* __restrict__ out, int E, int N) {
  long long gtid = (long long)blockIdx.x * blockDim.x + threadIdx.x;
  int wid  = (int)(gtid >> 5);         // uniform across the wave
  int lane = threadIdx.x & 31;
  int total = E + N;
  if (wid >= total) return;

  int row, col;
  if (wid < E) {
    row = (int)ei[wid];                // source
    col = (int)ei[E + wid];            // destination
  } else {
    row = col = wid - E;               // self loop
  }

  float nrm = dinv[row] * dinv[col];
  const float4* __restrict__ hp = (const float4*)(h + (size_t)row * D);
  float4 v = hp[lane];
  float* __restrict__ op = out + (size_t)col * D + lane * 4;
  atomicAdd(op + 0, v.x * nrm);
  atomicAdd(op + 1, v.y * nrm);
  atomicAdd(op + 2, v.z * nrm);
  atomicAdd(op + 3, v.w * nrm);
}

extern "C" void kernel_launch(void* const* d_in, const int* in_sizes, int n_in,
                              void* d_out, int out_size, void* d_ws, size_t ws_size,
                              hipStream_t stream) {
  const float*     x  = (const float*)d_in[0];       // [N, 128]
  const long long* ei = (const long long*)d_in[1];   // [2, E] int64
  const float*     W  = (const float*)d_in[2];       // [128, 128]
  const float*     b  = (const float*)d_in[3];       // [128]
  float* out = (float*)d_out;                        // [N, 128]

  const int N = in_sizes[0] / D;
  const int E = in_sizes[1] / 2;

  // Workspace: h [N*D floats] then deg/dinv [N floats] (~25.8 MB)
  float* h   = (float*)d_ws;
  float* deg = h + (size_t)N * D;

  {
    int threads = 256;
    int blocks = (N * D + threads - 1) / threads;
    gcn_init<<<blocks, threads, 0, stream>>>(out, deg, b, N);
  }
  {
    int threads = 256;
    int blocks = (E + threads - 1) / threads;
    gcn_degree<<<blocks, threads, 0, stream>>>(ei, deg, E);
  }
  {
    int threads = 256;
    int blocks = (N + threads - 1) / threads;
    gcn_rsqrt<<<blocks, threads, 0, stream>>>(deg, N);
  }
  {
    int threads = 256;                       // 8 waves -> 8 column tiles of 16
    int blocks = (N + 15) / 16;              // one 16-row stripe per block
    gcn_gemm<<<blocks, threads, 0, stream>>>(x, W, h, N);
  }
  {
    int totalWaves = E + N;
    int threads = 256;                       // 8 waves per block
    int blocks = (totalWaves + 7) / 8;
    gcn_scatter<<<blocks, threads, 0, stream>>>(ei, h, deg, out, E, N);
  }
}